// EP_SRNNAdapter_57080115363999
// MI455X (gfx1250) — compile-verified
//
#include <hip/hip_runtime.h>
#include <hip/hip_bf16.h>

typedef float v2f __attribute__((ext_vector_type(2)));
typedef float v8f __attribute__((ext_vector_type(8)));

#define B_    256
#define T_    100
#define D_    700
#define H_    1024
#define O_    20
#define STEPS (T_ - 1)                 /* 99 scan steps */
#define ALPHA 0.951229424500714f       /* exp(-1/20) */
#define KAPPA 0.951229424500714f
#define THRV  1.0f

// fp32 WMMA 16x16x4: D = A(16x4) * B(4x16) + C  (wave32)
static __device__ __forceinline__ v8f wmma_f32(v2f a, v2f b, v8f c) {
  return __builtin_amdgcn_wmma_f32_16x16x4_f32(
      false, a, false, b, (short)0, c, false, false);
}

// LDS layout for the A operand, stored in WMMA fragment order:
//   index = (k>>2)*64 + lane*2 + elem, lane = row + 16*((k&2)>>1), elem = k&1
// Inner loop A load = one linear, bank-conflict-free ds_load_b64 per lane.
static __device__ __forceinline__ int fragIdx(int r, int c) {
  return ((c >> 2) << 6) + ((r + ((c & 2) << 3)) << 1) + (c & 1);
}

// ---------------------------------------------------------------------------
// Init: Weff = Wrec with zeroed diagonal; zero V, Z0, VO, Out.
// ---------------------------------------------------------------------------
__global__ void srnn_init(const float* __restrict__ Wrec, float* __restrict__ Weff,
                          float* __restrict__ V, float* __restrict__ Z0,
                          float* __restrict__ VO, float* __restrict__ Out) {
  int i = blockIdx.x * blockDim.x + threadIdx.x;
  if (i < H_ * H_) {
    int r = i >> 10, c = i & (H_ - 1);
    Weff[i] = (r == c) ? 0.0f : Wrec[i];
  }
  if (i < B_ * H_) { V[i] = 0.0f; Z0[i] = 0.0f; }
  if (i < B_ * O_) { VO[i] = 0.0f; Out[i] = 0.0f; }
}

// ---------------------------------------------------------------------------
// Input projection: Xin[t, b, h] = sum_d X[b, t, d] * Win[h, d]
// Block = 128 threads (4 waves). Block stages 16 batch rows of x_t in LDS
// (fragment order); each wave computes a 16x64 tile (4 accumulators).
// grid = (4 h-groups, 16 btiles, 99 t).
// ---------------------------------------------------------------------------
__global__ __launch_bounds__(128) void srnn_in_gemm(const float* __restrict__ X,
                                                    const float* __restrict__ Win,
                                                    float* __restrict__ Xin) {
  __shared__ __align__(16) float lA[(D_ / 4) * 64];   // 11200 floats = 44.8 KB
  const int tid  = threadIdx.x;
  const int lane = tid & 31, wave = tid >> 5;
  const int bt = blockIdx.y, t = blockIdx.z;
  const int b0 = bt * 16;

  // Stage A = x_TBD[t, b0..b0+15, :] = X[b, t, :]  (coalesced global reads)
  for (int r = 0; r < 16; ++r) {
    const float* src = X + ((size_t)(b0 + r) * T_ + t) * D_;
    for (int c = tid; c < D_; c += 128)
      lA[fragIdx(r, c)] = src[c];
  }
  __syncthreads();

  const int row  = lane & 15;
  const int koff = (lane < 16) ? 0 : 2;
  const int h0   = (blockIdx.x * 4 + wave) * 64;
  const float* bbase = Win + (size_t)(h0 + row) * D_;

  v8f acc[4] = {};
  for (int kk = 0; kk < D_ / 4; ++kk) {
    v2f a = *(const v2f*)(lA + (kk << 6) + (lane << 1));
    const int k = (kk << 2) + koff;
#pragma unroll
    for (int j = 0; j < 4; ++j) {
      v2f b;
      b.x = bbase[(size_t)j * 16 * D_ + k];
      b.y = bbase[(size_t)j * 16 * D_ + k + 1];
      acc[j] = wmma_f32(a, b, acc[j]);
    }
  }

  const int mbase = (lane < 16) ? 0 : 8;
  const int n = lane & 15;
  float* obase = Xin + ((size_t)t * B_ + b0) * H_ + h0 + n;
#pragma unroll
  for (int j = 0; j < 4; ++j)
#pragma unroll
    for (int r = 0; r < 8; ++r)
      obase[(size_t)(mbase + r) * H_ + j * 16] = acc[j][r];
}

// ---------------------------------------------------------------------------
// One recurrent step:
//   v_new = alpha*v + Zin @ Weff^T + Xin[t] - THR*Zin ;  Zout = (v_new > THR)
// Block stages 16 z-rows (64 KB LDS, fragment order); each wave does a 16x32
// tile (2 accumulators). Zin/Zout ping-pong across launches.
// grid = (8 h-groups, 16 btiles).
// ---------------------------------------------------------------------------
__global__ __launch_bounds__(128) void srnn_rec_step(const float* __restrict__ Zin,
                                                     float* __restrict__ Zout,
                                                     float* __restrict__ V,
                                                     const float* __restrict__ Weff,
                                                     const float* __restrict__ Xin_t) {
  __shared__ __align__(16) float lZ[(H_ / 4) * 64];   // 16384 floats = 64 KB
  const int tid  = threadIdx.x;
  const int lane = tid & 31, wave = tid >> 5;
  const int b0 = blockIdx.y * 16;

  for (int r = 0; r < 16; ++r) {
    const float* src = Zin + (size_t)(b0 + r) * H_;
    for (int c = tid; c < H_; c += 128)
      lZ[fragIdx(r, c)] = src[c];
  }
  __syncthreads();

  const int row  = lane & 15;
  const int koff = (lane < 16) ? 0 : 2;
  const int h0   = (blockIdx.x * 4 + wave) * 32;
  const float* bbase = Weff + (size_t)(h0 + row) * H_;

  v8f acc[2] = {};
  for (int kk = 0; kk < H_ / 4; ++kk) {                // 256 iters
    v2f a = *(const v2f*)(lZ + (kk << 6) + (lane << 1));
    const int k = (kk << 2) + koff;
#pragma unroll
    for (int j = 0; j < 2; ++j) {
      v2f b;
      b.x = bbase[(size_t)j * 16 * H_ + k];
      b.y = bbase[(size_t)j * 16 * H_ + k + 1];
      acc[j] = wmma_f32(a, b, acc[j]);
    }
  }

  const int mbase = (lane < 16) ? 0 : 8;
  const int n = lane & 15;
#pragma unroll
  for (int j = 0; j < 2; ++j)
#pragma unroll
    for (int r = 0; r < 8; ++r) {
      const int m   = mbase + r;
      const int col = h0 + j * 16 + n;
      const size_t idx = (size_t)(b0 + m) * H_ + col;
      float zo   = lZ[fragIdx(m, col)];                // z_old from LDS
      float vnew = ALPHA * V[idx] + acc[j][r] + Xin_t[idx] - THRV * zo;
      V[idx]    = vnew;
      Zout[idx] = (vnew > THRV) ? 1.0f : 0.0f;
    }
}

// ---------------------------------------------------------------------------
// Output step: vo = kappa*vo + Znew @ Wout^T ; in last-10 window accumulate
// softmax(vo)/10 into Out. One wave per batch row, lane o<20 owns one class.
// ---------------------------------------------------------------------------
__global__ __launch_bounds__(128) void srnn_out_step(const float* __restrict__ Znew,
                                                     const float* __restrict__ Wout,
                                                     float* __restrict__ VO,
                                                     float* __restrict__ Out,
                                                     int accumulate) {
  const int lane = threadIdx.x & 31;
  const int wave = threadIdx.x >> 5;
  const int b = blockIdx.x * 4 + wave;        // 0..255

  const float* zrow = Znew + (size_t)b * H_;
  float acc = 0.0f;
  if (lane < O_) {
    const float* wrow = Wout + (size_t)lane * H_;
    float s = 0.0f;
    for (int j = 0; j < H_; ++j) s += zrow[j] * wrow[j];
    acc = KAPPA * VO[b * O_ + lane] + s;
    VO[b * O_ + lane] = acc;
  }
  if (accumulate) {
    float m = (lane < O_) ? acc : -__builtin_inff();
    for (int off = 16; off > 0; off >>= 1) m = fmaxf(m, __shfl_xor(m, off, 32));
    float e = (lane < O_) ? __expf(acc - m) : 0.0f;
    float s = e;
    for (int off = 16; off > 0; off >>= 1) s += __shfl_xor(s, off, 32);
    if (lane < O_) Out[b * O_ + lane] += (e / s) * 0.1f;
  }
}

// ---------------------------------------------------------------------------
extern "C" void kernel_launch(void* const* d_in, const int* in_sizes, int n_in,
                              void* d_out, int out_size, void* d_ws, size_t ws_size,
                              hipStream_t stream) {
  const float* X    = (const float*)d_in[0];   // [256,100,700]
  const float* Win  = (const float*)d_in[1];   // [1024,700]
  const float* Wrec = (const float*)d_in[2];   // [1024,1024]
  const float* Wout = (const float*)d_in[3];   // [20,1024]
  float* out = (float*)d_out;                  // [256,20]

  // Workspace layout (floats), ~111 MB total:
  float* ws   = (float*)d_ws;
  float* Xin  = ws;                                     // STEPS*B*H
  float* Weff = Xin  + (size_t)STEPS * B_ * H_;         // H*H
  float* V    = Weff + (size_t)H_ * H_;                 // B*H
  float* Za   = V    + (size_t)B_ * H_;                 // B*H  (z ping)
  float* Zb   = Za   + (size_t)B_ * H_;                 // B*H  (z pong)
  float* VO   = Zb   + (size_t)B_ * H_;                 // B*O

  srnn_init<<<(H_ * H_ + 255) / 256, 256, 0, stream>>>(Wrec, Weff, V, Za, VO, out);

  dim3 gA(4, 16, STEPS);
  srnn_in_gemm<<<gA, 128, 0, stream>>>(X, Win, Xin);

  float* zin = Za;
  float* zout = Zb;
  for (int t = 0; t < STEPS; ++t) {
    dim3 gB(8, 16);
    srnn_rec_step<<<gB, 128, 0, stream>>>(zin, zout, V, Weff,
                                          Xin + (size_t)t * B_ * H_);
    srnn_out_step<<<64, 128, 0, stream>>>(zout, Wout, VO, out,
                                          (t >= STEPS - 10) ? 1 : 0);
    float* tmp = zin; zin = zout; zout = tmp;
  }
}